// TrimSilence_72739566125614
// MI455X (gfx1250) — compile-verified
//
#include <hip/hip_runtime.h>

// ---- problem constants (from reference) ----
#define WIN   32000
#define HOP   4000
#define OFF   8000
#define TOPK  256
#define SEG   (2 * OFF)        // 16000
#define RWIN  (WIN / HOP)      // 8
#define NMAX  10240            // padded window count for LDS

typedef __attribute__((ext_vector_type(2))) float v2f;
typedef __attribute__((ext_vector_type(8))) float v8f;

#if __has_builtin(__builtin_amdgcn_wmma_f32_16x16x4_f32)
#define HAVE_WMMA_F32X4 1
#endif

// ------------------------------------------------------------------
// Kernel 1: per-HOP block sums of |x|. One wave (32 lanes) per chunk.
// WMMA trick: with B = ones(4x16), D = A*B + C accumulates exact f32
// sums of every A element into the C tile. 64 elements per v_wmma.
// ------------------------------------------------------------------
__global__ __launch_bounds__(256) void block_sums_kernel(
    const float* __restrict__ x, float* __restrict__ bs, int nblocks) {
  int wave = (int)((blockIdx.x * blockDim.x + threadIdx.x) >> 5);
  int lane = (int)(threadIdx.x & 31);
  if (wave >= nblocks) return;  // wave-uniform: EXEC stays all-ones for WMMA

  const float* p = x + (size_t)wave * HOP;

#if HAVE_WMMA_F32X4
  v8f acc = {};
  v2f ones; ones[0] = 1.0f; ones[1] = 1.0f;
  // 31 iterations * 128 floats (float4 per lane) = 3968, remainder 32.
  for (int it = 0; it < 31; ++it) {
    const float4* q = (const float4*)(p + it * 128) + lane;
    __builtin_prefetch((const char*)q + 8192, 0, 1);  // global_prefetch_b8
    float4 v = *q;
    v2f a0; a0[0] = fabsf(v.x); a0[1] = fabsf(v.y);
    v2f a1; a1[0] = fabsf(v.z); a1[1] = fabsf(v.w);
    acc = __builtin_amdgcn_wmma_f32_16x16x4_f32(false, a0, false, ones,
                                                (short)0, acc, false, false);
    acc = __builtin_amdgcn_wmma_f32_16x16x4_f32(false, a1, false, ones,
                                                (short)0, acc, false, false);
  }
  {  // tail: one float per lane, pad K-slot with 0
    float v = p[31 * 128 + lane];
    v2f a0; a0[0] = fabsf(v); a0[1] = 0.0f;
    acc = __builtin_amdgcn_wmma_f32_16x16x4_f32(false, a0, false, ones,
                                                (short)0, acc, false, false);
  }
  // C/D layout: lane L<16 holds rows M=0..7 at N=L; lane L>=16 rows M=8..15.
  float s = acc[0] + acc[1] + acc[2] + acc[3] + acc[4] + acc[5] + acc[6] + acc[7];
  float total = __shfl(s, 0, 32) + __shfl(s, 16, 32);  // column N=0, all 16 rows
#else
  float s = 0.0f;
  for (int it = 0; it < 31; ++it) {
    const float4* q = (const float4*)(p + it * 128) + lane;
    __builtin_prefetch((const char*)q + 8192, 0, 1);
    float4 v = *q;
    s += fabsf(v.x) + fabsf(v.y) + fabsf(v.z) + fabsf(v.w);
  }
  s += fabsf(p[31 * 128 + lane]);
  #pragma unroll
  for (int o = 16; o > 0; o >>= 1) s += __shfl_xor(s, o, 32);
  float total = s;
#endif
  if (lane == 0) bs[wave] = total;
}

// ------------------------------------------------------------------
// Kernel 2: sliding-window means over 8 block sums.
// ------------------------------------------------------------------
__global__ __launch_bounds__(256) void means_kernel(
    const float* __restrict__ bs, float* __restrict__ means, int nwin) {
  int w = (int)(blockIdx.x * blockDim.x + threadIdx.x);
  if (w >= nwin) return;
  float s = 0.0f;
  #pragma unroll
  for (int r = 0; r < RWIN; ++r) s += bs[w + r];
  means[w] = s * (1.0f / (float)WIN);
}

// ------------------------------------------------------------------
// Kernel 3: rank every window against all others (top_k semantics:
// larger mean wins; ties broken by smaller index). Each block caches
// all means in LDS (40KB of the 320KB WGP pool), broadcast b128 reads.
// ------------------------------------------------------------------
__global__ __launch_bounds__(256) void rank_kernel(
    const float* __restrict__ means, int nwin, unsigned int* __restrict__ sel) {
  __shared__ __align__(16) float sm[NMAX];
  for (int i = (int)threadIdx.x; i < NMAX; i += 256)
    sm[i] = (i < nwin) ? means[i] : -1.0e30f;   // sentinel: never selected, never ties
  __syncthreads();

  int w = (int)(blockIdx.x * 256 + threadIdx.x);
  if (w >= nwin) return;
  float mw = sm[w];
  int n4 = (nwin + 3) & ~3;
  int rank = 0;
  for (int v = 0; v < n4; v += 4) {
    float4 q = *(const float4*)&sm[v];   // LDS broadcast across the wave
    rank += (q.x > mw) || (q.x == mw && (v + 0) < w);
    rank += (q.y > mw) || (q.y == mw && (v + 1) < w);
    rank += (q.z > mw) || (q.z == mw && (v + 2) < w);
    rank += (q.w > mw) || (q.w == mw && (v + 3) < w);
  }
  sel[w] = (rank < TOPK) ? 1u : 0u;
}

// ------------------------------------------------------------------
// Kernel 4: compact selected windows into sorted centers (block scan).
// ------------------------------------------------------------------
__global__ __launch_bounds__(1024) void compact_kernel(
    const unsigned int* __restrict__ sel, int nwin, int* __restrict__ centers) {
  __shared__ int cnt[1024];
  int t = (int)threadIdx.x;
  int per = (nwin + 1023) >> 10;
  int begin = t * per;
  int end = begin + per; if (end > nwin) end = nwin;
  if (begin > nwin) begin = nwin;

  int c = 0;
  for (int w = begin; w < end; ++w) c += (int)sel[w];
  cnt[t] = c;
  __syncthreads();
  for (int off = 1; off < 1024; off <<= 1) {  // Hillis-Steele inclusive scan
    int v = (t >= off) ? cnt[t - off] : 0;
    __syncthreads();
    cnt[t] += v;
    __syncthreads();
  }
  int slot = cnt[t] - c;  // exclusive prefix
  for (int w = begin; w < end; ++w) {
    if (sel[w]) {
      if (slot < TOPK) centers[slot] = w * HOP + WIN / 2;
      ++slot;
    }
  }
}

// ------------------------------------------------------------------
// Kernel 5: gather 256 segments, apply fade mask t*(1-t), write out.
// Segment starts are multiples of 4000 floats -> 16B aligned.
// ------------------------------------------------------------------
__global__ __launch_bounds__(256) void gather_kernel(
    const float* __restrict__ x, const int* __restrict__ centers,
    float* __restrict__ out) {
  int k = (int)blockIdx.x;
  int c = centers[k];
  const float4* __restrict__ src = (const float4*)(x + (size_t)(c - OFF));
  float4* __restrict__ dst = (float4*)(out + (size_t)k * SEG);
  const float inv = 1.0f / (float)(SEG - 1);
  for (int i = (int)threadIdx.x; i < SEG / 4; i += 256) {
    __builtin_prefetch((const char*)(src + i) + 8192, 0, 1);
    float4 v = src[i];
    int j = i * 4;
    float t0 = (float)(j + 0) * inv, t1 = (float)(j + 1) * inv;
    float t2 = (float)(j + 2) * inv, t3 = (float)(j + 3) * inv;
    v.x *= t0 * (1.0f - t0);
    v.y *= t1 * (1.0f - t1);
    v.z *= t2 * (1.0f - t2);
    v.w *= t3 * (1.0f - t3);
    dst[i] = v;
  }
}

// ------------------------------------------------------------------
extern "C" void kernel_launch(void* const* d_in, const int* in_sizes, int n_in,
                              void* d_out, int out_size, void* d_ws, size_t ws_size,
                              hipStream_t stream) {
  (void)n_in; (void)out_size; (void)ws_size;
  const float* x = (const float*)d_in[0];
  const int T = in_sizes[0];                 // 40,000,000
  const int NB = T / HOP;                    // 10,000
  const int NWIN = (T - WIN) / HOP + 1;      // 9,993

  float* bs = (float*)d_ws;                          // NB floats
  float* means = bs + NB;                            // NWIN floats
  unsigned int* sel = (unsigned int*)(means + NB);   // NWIN u32 (NB-padded region)
  int* centers = (int*)(sel + NB);                   // TOPK ints

  // 1 wave per HOP chunk, 8 chunks per 256-thread block
  block_sums_kernel<<<(NB + 7) / 8, 256, 0, stream>>>(x, bs, NB);
  means_kernel<<<(NWIN + 255) / 256, 256, 0, stream>>>(bs, means, NWIN);
  rank_kernel<<<(NWIN + 255) / 256, 256, 0, stream>>>(means, NWIN, sel);
  compact_kernel<<<1, 1024, 0, stream>>>(sel, NWIN, centers);
  gather_kernel<<<TOPK, 256, 0, stream>>>(x, centers, (float*)d_out);
}